// Quan2d_19980187861747
// MI455X (gfx1250) — compile-verified
//
#include <hip/hip_runtime.h>
#include <hip/hip_bf16.h>

typedef __attribute__((ext_vector_type(2))) float v2f;
typedef __attribute__((ext_vector_type(8))) float v8f;

// ---------------------------------------------------------------------------
// Setup: collapse the fixed CRZ+RY layer + measurement into W[16][16]
// (only columns 0..3 nonzero).  ev[q] = sum_k f[k] * W[k][q].
// Features: k=0..3  -> |w_i|^2
//           k=4..9  -> Re(conj(w_i) w_j) for pairs (01,02,03,12,13,23)
//           k=10..15-> Im(conj(w_i) w_j) same pairs
// All normalization (1/16) folded into W.
// ---------------------------------------------------------------------------
__global__ void quanv_setup(const float* __restrict__ wts, float* __restrict__ W)
{
    if (threadIdx.x != 0 || blockIdx.x != 0) return;

    float th[4], cb[4], sb[4];
    for (int k = 0; k < 4; ++k) th[k] = wts[k];
    for (int k = 0; k < 4; ++k) { cb[k] = cosf(wts[4 + k]); sb[k] = sinf(wts[4 + k]); }

    // D[m] = product of CRZ phases; basis m = b0*8 + b1*4 + b2*2 + b3
    float Dre[16], Dim[16];
    for (int m = 0; m < 16; ++m) {
        float ph = 0.f;
        for (int i = 0; i < 4; ++i) {
            int bc = (m >> (3 - i)) & 1;
            int tq = (i + 1) & 3;
            int bt = (m >> (3 - tq)) & 1;
            if (bc) ph += (bt ? 0.5f : -0.5f) * th[i];
        }
        Dre[m] = cosf(ph); Dim[m] = sinf(ph);
    }

    // A_q[i][j] = (1/16) * sum_{bb,cc} conj(D[m]) * G_q[m,n] * D[n]
    // with m=i*4+bb, n=j*4+cc; G_q factorizes: delta on bits k!=q,
    // and g_q = [[cosB, -sinB], [-sinB, -cosB]] on bit q.
    float Are[4][4][4], Aim[4][4][4];
    for (int q = 0; q < 4; ++q)
        for (int i = 0; i < 4; ++i)
            for (int j = 0; j < 4; ++j) {
                float ar = 0.f, ai = 0.f;
                for (int bb = 0; bb < 4; ++bb)
                    for (int cc = 0; cc < 4; ++cc) {
                        int m = i * 4 + bb, n = j * 4 + cc;
                        bool ok = true;
                        for (int k = 0; k < 4; ++k)
                            if (k != q && (((m >> (3 - k)) & 1) != ((n >> (3 - k)) & 1)))
                                ok = false;
                        if (!ok) continue;
                        int bm = (m >> (3 - q)) & 1, bn = (n >> (3 - q)) & 1;
                        float g = (bm == bn) ? (bm ? -cb[q] : cb[q]) : -sb[q];
                        ar += (Dre[m] * Dre[n] + Dim[m] * Dim[n]) * g;
                        ai += (Dre[m] * Dim[n] - Dim[m] * Dre[n]) * g;
                    }
                Are[q][i][j] = ar * 0.0625f;
                Aim[q][i][j] = ai * 0.0625f;
            }

    for (int t = 0; t < 256; ++t) W[t] = 0.f;
    const int PIdx[6] = {0, 0, 0, 1, 1, 2};
    const int PJdx[6] = {1, 2, 3, 2, 3, 3};
    for (int q = 0; q < 4; ++q) {
        for (int i = 0; i < 4; ++i) W[i * 16 + q] = Are[q][i][i];
        for (int p = 0; p < 6; ++p) {
            W[(4 + p)  * 16 + q] =  2.f * Are[q][PIdx[p]][PJdx[p]];
            W[(10 + p) * 16 + q] = -2.f * Aim[q][PIdx[p]][PJdx[p]];
        }
    }
}

// ---------------------------------------------------------------------------
// Main: one thread per 2x2 patch.  Per wave (32 patches): compute 16 features,
// stage through LDS into WMMA A-layout, contract with W via 4 chained
// v_wmma_f32_16x16x4_f32 (K=16) per 16-patch tile, write 4 expectation values.
// ---------------------------------------------------------------------------
__global__ __launch_bounds__(256) void quanv_main(const float* __restrict__ x,
                                                  const float* __restrict__ W,
                                                  float* __restrict__ out)
{
    __shared__ float lds[8 * 544];           // per wave: 32 patches * 17-stride
    const int lane = threadIdx.x & 31;
    const int wv   = threadIdx.x >> 5;
    const int p    = blockIdx.x * 256 + threadIdx.x;

    // patch p -> (b, r, c) for x(B,1,64,64); patch grid 32x32
    const int b    = p >> 10;
    const int rc   = p & 1023;
    const int r    = rc >> 5;
    const int c    = rc & 31;
    const int base = (b << 12) + (r << 7) + (c << 1);

    const float2 t0 = *(const float2*)(x + base);        // row 2r:   a0, a1
    const float2 t1 = *(const float2*)(x + base + 64);   // row 2r+1: a2, a3

    // phi = RY(ay) RZ(az) |+>, unnormalized (1/sqrt2 folded into W)
    float sz0, cz0, sy0, cy0, sz1, cz1, sy1, cy1;
    __sincosf(0.5f * t0.x, &sz0, &cz0);
    __sincosf(0.5f * t0.y, &sy0, &cy0);
    __sincosf(0.5f * t1.x, &sz1, &cz1);
    __sincosf(0.5f * t1.y, &sy1, &cy1);

    float p0r[2], p0i[2], p1r[2], p1i[2];
    p0r[0] = (cy0 - sy0) * cz0;  p0i[0] = -(cy0 + sy0) * sz0;
    p0r[1] = (cy0 + sy0) * cz0;  p0i[1] =  (cy0 - sy0) * sz0;
    p1r[0] = (cy1 - sy1) * cz1;  p1i[0] = -(cy1 + sy1) * sz1;
    p1r[1] = (cy1 + sy1) * cz1;  p1i[1] =  (cy1 - sy1) * sz1;

    float wr[4], wi[4];
#pragma unroll
    for (int i0 = 0; i0 < 2; ++i0)
#pragma unroll
        for (int i1 = 0; i1 < 2; ++i1) {
            int i = i0 * 2 + i1;
            wr[i] = p0r[i0] * p1r[i1] - p0i[i0] * p1i[i1];
            wi[i] = p0r[i0] * p1i[i1] + p0i[i0] * p1r[i1];
        }

    float f[16];
#pragma unroll
    for (int i = 0; i < 4; ++i) f[i] = wr[i] * wr[i] + wi[i] * wi[i];
    const int PIdx[6] = {0, 0, 0, 1, 1, 2};
    const int PJdx[6] = {1, 2, 3, 2, 3, 3};
#pragma unroll
    for (int pp = 0; pp < 6; ++pp) {
        int i = PIdx[pp], j = PJdx[pp];
        f[4 + pp]  = wr[i] * wr[j] + wi[i] * wi[j];
        f[10 + pp] = wr[i] * wi[j] - wi[i] * wr[j];
    }

    float* fl = &lds[wv * 544];
#pragma unroll
    for (int k = 0; k < 16; ++k) fl[lane * 17 + k] = f[k];   // stride 17: conflict-free

    __syncthreads();

    const int nq  = lane & 15;           // N index (q) / source patch row M
    const int hi2 = (lane >> 4) << 1;    // lanes 16-31 carry K+2,K+3

    // B matrix (16x16, K x N), VGPR0 = rows K / K+2, VGPR1 = rows K+1 / K+3
    v2f bm[4];
#pragma unroll
    for (int v = 0; v < 4; ++v) {
        bm[v].x = W[(4 * v + hi2)     * 16 + nq];
        bm[v].y = W[(4 * v + 1 + hi2) * 16 + nq];
    }

    const int waveBase = blockIdx.x * 256 + wv * 32;
#pragma unroll
    for (int t = 0; t < 2; ++t) {
        const int sp = t * 16 + nq;      // within-wave patch supplying row M
        v8f acc = {};
#pragma unroll
        for (int v = 0; v < 4; ++v) {
            v2f av;
            av.x = fl[sp * 17 + 4 * v + hi2];
            av.y = fl[sp * 17 + 4 * v + 1 + hi2];
            acc = __builtin_amdgcn_wmma_f32_16x16x4_f32(
                      false, av, false, bm[v], (short)0, acc, false, false);
        }
        // D layout: VGPR rr -> M = rr (lanes 0-15) / 8+rr (lanes 16-31), N = lane%16
        const int mhi = (lane >> 4) * 8;
        if (nq < 4) {
#pragma unroll
            for (int rr = 0; rr < 8; ++rr) {
                int patch = waveBase + t * 16 + mhi + rr;
                out[patch * 4 + nq] = acc[rr];
            }
        }
    }
}

extern "C" void kernel_launch(void* const* d_in, const int* in_sizes, int n_in,
                              void* d_out, int out_size, void* d_ws, size_t ws_size,
                              hipStream_t stream)
{
    const float* x   = (const float*)d_in[0];   // (512,1,64,64) f32
    const float* wts = (const float*)d_in[1];   // (1,8) f32
    float* out = (float*)d_out;                  // (512,4096) f32
    float* W   = (float*)d_ws;                   // 256 floats scratch

    const int total  = in_sizes[0];              // B*64*64
    const int npatch = total >> 2;               // B*32*32  (multiple of 256)

    quanv_setup<<<1, 32, 0, stream>>>(wts, W);
    quanv_main<<<npatch / 256, 256, 0, stream>>>(x, W, out);
}